// CriticGNN_39779987095910
// MI455X (gfx1250) — compile-verified
//
#include <hip/hip_runtime.h>

typedef __attribute__((ext_vector_type(2))) float v2f;
typedef __attribute__((ext_vector_type(8))) float v8f;
typedef __attribute__((ext_vector_type(4))) unsigned int u32x4;
typedef __attribute__((ext_vector_type(8))) int i32x8;
typedef __attribute__((ext_vector_type(4))) int i32x4;

#define HID  128
#define INF  32

// ---------------- degree / norm ----------------
__global__ void k_init_deg(float* __restrict__ deg, int N) {
    int i = blockIdx.x * blockDim.x + threadIdx.x;
    if (i < N) deg[i] = 1.0f;                       // implicit self loop
}

__global__ void k_deg_accum(const int* __restrict__ col, float* __restrict__ deg, int E) {
    int e = blockIdx.x * blockDim.x + threadIdx.x;
    if (e < E) unsafeAtomicAdd(&deg[col[e]], 1.0f);
}

__global__ void k_edge_norm(const int* __restrict__ row, const int* __restrict__ col,
                            const float* __restrict__ deg, float* __restrict__ norm, int E) {
    int e = blockIdx.x * blockDim.x + threadIdx.x;
    if (e < E) norm[e] = rsqrtf(deg[row[e]]) * rsqrtf(deg[col[e]]);
}

// ---------------- embed: h = relu(x @ emb_W + emb_b), [N,32]x[32,128] ----------------
__global__ void k_embed_wmma(const float* __restrict__ x, const float* __restrict__ W,
                             const float* __restrict__ b, float* __restrict__ h, int N) {
    __shared__ float sA[16 * INF];                  // 16-row x tile
    const int m0 = blockIdx.x * 16;
    const int t  = threadIdx.x;                     // 0..255, 8 waves

    if (t < INF) __builtin_prefetch(W + t * HID, 0, 1);   // global_prefetch_b8

    for (int i = t; i < 16 * INF; i += 256) {
        int gi = m0 * INF + i;
        sA[i] = (gi < N * INF) ? x[gi] : 0.0f;
    }
    __syncthreads();

    const int wave = t >> 5, lane = t & 31;
    const int half = lane >> 4, r = lane & 15;
    const int n0 = wave * 16;

    v8f c = {};
    #pragma unroll
    for (int kk = 0; kk < INF / 4; ++kk) {
        const int K = kk * 4;
        v2f a, bv;
        a.x  = sA[r * INF + K + (half ? 2 : 0)];
        a.y  = sA[r * INF + K + (half ? 3 : 1)];
        bv.x = W[(K + (half ? 2 : 0)) * HID + n0 + r];
        bv.y = W[(K + (half ? 3 : 1)) * HID + n0 + r];
        c = __builtin_amdgcn_wmma_f32_16x16x4_f32(false, a, false, bv,
                                                  (short)0, c, false, false);
    }
    const float bias = b[n0 + r];
    #pragma unroll
    for (int v = 0; v < 8; ++v) {
        int m = m0 + v + 8 * half;
        if (m < N) h[(size_t)m * HID + n0 + r] = fmaxf(c[v] + bias, 0.0f);
    }
}

// ---------------- conv GEMM (TDM-staged A tile):
//   hw  = h @ W                       [N,128]x[128,128]
//   agg = hw * (1/deg)                (fused self-loop init)
__global__ void k_conv_wmma(const float* __restrict__ h, const float* __restrict__ W,
                            const float* __restrict__ deg,
                            float* __restrict__ hw, float* __restrict__ agg, int N) {
    __shared__ float sA[16 * HID];                  // 16-row h tile (8 KB)
    const int m0 = blockIdx.x * 16;
    const int t  = threadIdx.x;

    if (t < HID) __builtin_prefetch(W + t * HID, 0, 1);   // global_prefetch_b8

    // Tensor Data Mover: DMA the 16x128 f32 tile from h into LDS.
    // D# per CDNA5 ISA ch.8; rows beyond tensor_dim1=N read back as zero.
    if (t == 0) {
        unsigned lds_addr = (unsigned)(size_t)(&sA[0]);
        unsigned long long ga =
            (unsigned long long)(size_t)h + (unsigned long long)m0 * HID * 4ull;
        u32x4 g0;
        g0[0] = 1u;                                       // count=1, user mode
        g0[1] = lds_addr;                                 // lds_addr [63:32]
        g0[2] = (unsigned)(ga & 0xFFFFFFFFull);           // global_addr lo
        g0[3] = (unsigned)((ga >> 32) & 0x01FFFFFFull)    // global_addr hi (bits 56:32)
                | (2u << 30);                             // type = 2 ("image")
        i32x8 g1;
        g1[0] = (2 << 16);                                // data_size = 2 (4 bytes)
        g1[1] = (HID << 16);                              // tensor_dim0[15:0] = 128
        g1[2] = (N & 0xFFFF) << 16;                       // tensor_dim1 lo
        g1[3] = ((N >> 16) & 0xFFFF) | (HID << 16);       // tensor_dim1 hi | tile_dim0=128
        g1[4] = 16;                                       // tile_dim1 = 16 rows
        g1[5] = HID;                                      // tensor_dim0_stride = 128
        g1[6] = 0;
        g1[7] = 0;
        i32x4 gz4 = {};                                   // groups 2/3 unused (2-D)
        i32x8 gz8 = {};                                   // extra group (clang-23 6-arg form)
        __builtin_amdgcn_tensor_load_to_lds(g0, g1, gz4, gz4, gz8, 0);
        __builtin_amdgcn_s_wait_tensorcnt(0);
    }
    __syncthreads();

    const int wave = t >> 5, lane = t & 31;
    const int half = lane >> 4, r = lane & 15;
    const int n0 = wave * 16;

    v8f c = {};
    #pragma unroll 8
    for (int kk = 0; kk < HID / 4; ++kk) {
        const int K = kk * 4;
        v2f a, bv;
        a.x  = sA[r * HID + K + (half ? 2 : 0)];
        a.y  = sA[r * HID + K + (half ? 3 : 1)];
        bv.x = W[(K + (half ? 2 : 0)) * HID + n0 + r];
        bv.y = W[(K + (half ? 3 : 1)) * HID + n0 + r];
        c = __builtin_amdgcn_wmma_f32_16x16x4_f32(false, a, false, bv,
                                                  (short)0, c, false, false);
    }
    #pragma unroll
    for (int v = 0; v < 8; ++v) {
        int m = m0 + v + 8 * half;
        if (m < N) {
            float val = c[v];
            size_t idx = (size_t)m * HID + n0 + r;
            hw[idx]  = val;
            agg[idx] = val * (1.0f / deg[m]);    // self-loop term seeds agg
        }
    }
}

// ---------------- edge scatter: agg[col] += hw[row] * norm[e] ----------------
__global__ void k_scatter(const int* __restrict__ row, const int* __restrict__ col,
                          const float* __restrict__ norm, const float4* __restrict__ hw4,
                          float* __restrict__ agg, int E) {
    int tid = blockIdx.x * blockDim.x + threadIdx.x;    // E * 32 threads, 4 floats each
    if (tid >= E * (HID / 4)) return;
    int e = tid >> 5;
    int c = tid & 31;
    int rsrc = row[e], dst = col[e];
    float w = norm[e];
    float4 v = hw4[(size_t)rsrc * (HID / 4) + c];
    float* a = agg + (size_t)dst * HID + c * 4;
    unsafeAtomicAdd(a + 0, v.x * w);
    unsafeAtomicAdd(a + 1, v.y * w);
    unsafeAtomicAdd(a + 2, v.z * w);
    unsafeAtomicAdd(a + 3, v.w * w);
}

// ---------------- finalize: h = relu(agg + conv_b); optional fused pooling ----------------
__global__ void k_finalize(const float* __restrict__ agg, const float* __restrict__ b,
                           float* __restrict__ h, const int* __restrict__ batch,
                           float* __restrict__ sums, int N, int do_pool) {
    int i = blockIdx.x * blockDim.x + threadIdx.x;      // over N * 128
    if (i >= N * HID) return;
    float val = fmaxf(agg[i] + b[i & (HID - 1)], 0.0f);
    h[i] = val;
    if (do_pool) {
        int n = i >> 7;
        int g = batch[n];
        unsafeAtomicAdd(&sums[(size_t)g * HID + (i & (HID - 1))], val);
    }
}

// ---------------- mean pooling setup ----------------
__global__ void k_pool_zero(float* __restrict__ sums, float* __restrict__ cnt, int G) {
    int i = blockIdx.x * blockDim.x + threadIdx.x;
    if (i < G * HID) sums[i] = 0.0f;
    if (i < G) cnt[i] = 0.0f;
}

__global__ void k_cnt_accum(const int* __restrict__ batch, float* __restrict__ cnt, int N) {
    int i = blockIdx.x * blockDim.x + threadIdx.x;
    if (i < N) unsafeAtomicAdd(&cnt[batch[i]], 1.0f);
}

// ---------------- head: out[g] = (sums[g] @ head_W) / max(cnt,1) + head_b ----------------
__global__ void k_head(const float* __restrict__ sums, const float* __restrict__ cnt,
                       const float* __restrict__ hW, const float* __restrict__ hb,
                       float* __restrict__ out, int G) {
    int g = blockIdx.x * blockDim.x + threadIdx.x;
    if (g >= G) return;
    const float* s = sums + (size_t)g * HID;
    float acc = 0.0f;
    #pragma unroll 8
    for (int f = 0; f < HID; ++f) acc += s[f] * hW[f];
    out[g] = acc / fmaxf(cnt[g], 1.0f) + hb[0];
}

extern "C" void kernel_launch(void* const* d_in, const int* in_sizes, int n_in,
                              void* d_out, int out_size, void* d_ws, size_t ws_size,
                              hipStream_t stream) {
    const float* x      = (const float*)d_in[0];
    const int*   ei     = (const int*)  d_in[1];   // [2,E] flat: row then col
    const int*   batch  = (const int*)  d_in[2];
    const float* emb_W  = (const float*)d_in[4];
    const float* emb_b  = (const float*)d_in[5];
    const float* conv_W = (const float*)d_in[6];
    const float* conv_b = (const float*)d_in[7];
    const float* head_W = (const float*)d_in[8];
    const float* head_b = (const float*)d_in[9];
    float* out = (float*)d_out;

    const int N = in_sizes[0] / INF;
    const int E = in_sizes[1] / 2;
    const int G = out_size;
    const int* row = ei;
    const int* col = ei + E;

    // workspace carve (256B aligned)
    char* p = (char*)d_ws;
    auto carve = [&](size_t bytes) -> float* {
        char* r = p;
        p += (bytes + 255) & ~(size_t)255;
        return (float*)r;
    };
    float* deg  = carve((size_t)N * 4);
    float* norm = carve((size_t)E * 4);
    float* h    = carve((size_t)N * HID * 4);
    float* hw   = carve((size_t)N * HID * 4);
    float* agg  = carve((size_t)N * HID * 4);
    float* sums = carve((size_t)G * HID * 4);
    float* cnt  = carve((size_t)G * 4);

    const int B = 256;
    auto cdiv = [](long long a, long long b) { return (int)((a + b - 1) / b); };

    // degrees + edge norms
    k_init_deg <<<cdiv(N, B), B, 0, stream>>>(deg, N);
    k_deg_accum<<<cdiv(E, B), B, 0, stream>>>(col, deg, E);
    k_edge_norm<<<cdiv(E, B), B, 0, stream>>>(row, col, deg, norm, E);

    // pooling buffers (zeroed before the fused last-layer accumulation)
    k_pool_zero<<<cdiv((long long)G * HID, B), B, 0, stream>>>(sums, cnt, G);
    k_cnt_accum<<<cdiv(N, B), B, 0, stream>>>(batch, cnt, N);

    // embedding GEMM (WMMA)
    k_embed_wmma<<<cdiv(N, 16), B, 0, stream>>>(x, emb_W, emb_b, h, N);

    // 3 GCN layers
    for (int l = 0; l < 3; ++l) {
        k_conv_wmma<<<cdiv(N, 16), B, 0, stream>>>(
            h, conv_W + (size_t)l * HID * HID, deg, hw, agg, N);
        k_scatter  <<<cdiv((long long)E * (HID / 4), B), B, 0, stream>>>(
            row, col, norm, (const float4*)hw, agg, E);
        k_finalize <<<cdiv((long long)N * HID, B), B, 0, stream>>>(
            agg, conv_b + (size_t)l * HID, h, batch, sums, N, (l == 2) ? 1 : 0);
    }

    // head on pooled sums
    k_head<<<cdiv(G, B), B, 0, stream>>>(sums, cnt, head_W, head_b, out, G);
}